// GraphEncoder_52441550684713
// MI455X (gfx1250) — compile-verified
//
#include <hip/hip_runtime.h>
#include <math.h>

#define NEG_SLOPE 0.2f

typedef float v2f __attribute__((ext_vector_type(2)));
typedef float v8f __attribute__((ext_vector_type(8)));

// Order-preserving float<->uint encoding so atomicMax(uint) == float max.
__device__ __forceinline__ unsigned enc_f(float f) {
  unsigned u = __float_as_uint(f);
  return (u & 0x80000000u) ? ~u : (u | 0x80000000u);
}
__device__ __forceinline__ float dec_f(unsigned u) {
  return (u & 0x80000000u) ? __uint_as_float(u & 0x7fffffffu)
                           : __uint_as_float(~u);
}

// ---------------------------------------------------------------------------
// GEMM: Y[M x (NT*16)] = X[M x 128] @ W[128 x (NT*16)] + bias, fp32 WMMA.
// W is staged into LDS once per block via CDNA5 async global->LDS copies
// (ASYNCcnt / s_wait_asynccnt), then all 8 waves serve B fragments from LDS.
// One wave per 16-row tile; NT column tiles accumulated in registers.
// A frag (16x4 f32): lane<16 -> M=lane, K={k0,k0+1}; lane>=16 -> K={k0+2,k0+3}
// B frag (4x16 f32): vgpr j, lanes0-15 -> K=k0+j (N=lane), lanes16-31 -> K=k0+2+j
// C/D (16x16 f32):   vgpr v, lanes0-15 -> M=v, N=lane; lanes16-31 -> M=v+8
// ---------------------------------------------------------------------------
template <int NT>
__global__ __launch_bounds__(256) void gemm_k128_wmma(
    const float* __restrict__ X, const float* __restrict__ W,
    const float* __restrict__ bias, float* __restrict__ Y, int M) {
  const int NC = NT * 16;
  __shared__ float shW[128 * NT * 16];  // 128 x NC, row-major (NT=8 -> 64 KB)

  // ---- Stage W into LDS: 128*NC*4 bytes = 32*NC 16B-chunks, 2*NT per thread.
  {
    unsigned lbase = (unsigned)(size_t)&shW[0];  // low 32 bits = LDS offset
    unsigned long long wbase = (unsigned long long)(size_t)W;
#pragma unroll
    for (int i = 0; i < 2 * NT; ++i) {
      unsigned off = (unsigned)((i * 256 + threadIdx.x) * 16);
      unsigned laddr = lbase + off;
      asm volatile("global_load_async_to_lds_b128 %0, %1, %2"
                   :
                   : "v"(laddr), "v"(off), "s"(wbase)
                   : "memory");
    }
    asm volatile("s_wait_asynccnt 0x0" ::: "memory");
  }
  __syncthreads();

  int wave = threadIdx.x >> 5;
  int lane = threadIdx.x & 31;
  int rt = blockIdx.x * 8 + wave;  // 16-row tile index (wave-uniform)
  if (rt * 16 >= M) return;
  int row0 = rt * 16;
  int half = lane >> 4;
  int mr = lane & 15;

  v8f acc[NT];
#pragma unroll
  for (int t = 0; t < NT; ++t) acc[t] = {};

  const float* xrow = X + (size_t)(row0 + mr) * 128;
  for (int k0 = 0; k0 < 128; k0 += 4) {
    v2f a;
    a.x = xrow[k0 + 2 * half + 0];
    a.y = xrow[k0 + 2 * half + 1];
    const float* w0 = &shW[(k0 + 2 * half) * NC + mr];
#pragma unroll
    for (int t = 0; t < NT; ++t) {
      v2f b;
      b.x = w0[t * 16];
      b.y = w0[NC + t * 16];
      acc[t] = __builtin_amdgcn_wmma_f32_16x16x4_f32(
          false, a, false, b, (short)0, acc[t], false, false);
    }
  }
#pragma unroll
  for (int t = 0; t < NT; ++t) {
    float bv = bias[t * 16 + mr];
#pragma unroll
    for (int v = 0; v < 8; ++v) {
      Y[(size_t)(row0 + v + 8 * half) * NC + t * 16 + mr] = acc[t][v] + bv;
    }
  }
}

// ---------------------------------------------------------------------------
// Self-loop edge_attr mean:  sums/cnts per dst over non-self edges.
// ---------------------------------------------------------------------------
__global__ void zero2_kernel(float* a, float* b, int n) {
  int i = blockIdx.x * blockDim.x + threadIdx.x;
  if (i < n) { a[i] = 0.f; b[i] = 0.f; }
}

__global__ void loopattr_edge_kernel(const long long* __restrict__ ei,
                                     const float* __restrict__ ea,
                                     float* sums, float* cnts, int E) {
  int e = blockIdx.x * blockDim.x + threadIdx.x;
  if (e >= E) return;
  int s = (int)ei[e], d = (int)ei[(size_t)E + e];
  if (s != d) {
    atomicAdd(&sums[d], ea[e]);
    atomicAdd(&cnts[d], 1.0f);
  }
}

// ---------------------------------------------------------------------------
// Node self-loop score: one wave per node. lane covers hc = lane*4 .. lane*4+3,
// head = lane/8.  Writes score_loop and seeds smax (encoded uint).
// ---------------------------------------------------------------------------
__global__ __launch_bounds__(256) void node_score_kernel(
    const float* __restrict__ xl, const float* __restrict__ xr,
    const float* __restrict__ We, const float* __restrict__ att,
    const float* __restrict__ sums, const float* __restrict__ cnts,
    float* __restrict__ sloop, unsigned* __restrict__ smax_u, int N) {
  int n = (int)((blockIdx.x * (size_t)blockDim.x + threadIdx.x) >> 5);
  int lane = threadIdx.x & 31;
  if (n >= N) return;
  float cnt = cnts[n];
  float la = cnt > 0.f ? sums[n] / cnt : 0.f;
  int base = lane * 4;
  float4 xls = *(const float4*)(xl + (size_t)n * 128 + base);
  float4 xrd = *(const float4*)(xr + (size_t)n * 128 + base);
  float4 we = *(const float4*)(We + base);
  float4 at = *(const float4*)(att + base);
  float m, sum = 0.f;
  m = xls.x + xrd.x + la * we.x; m = m > 0.f ? m : NEG_SLOPE * m; sum += at.x * m;
  m = xls.y + xrd.y + la * we.y; m = m > 0.f ? m : NEG_SLOPE * m; sum += at.y * m;
  m = xls.z + xrd.z + la * we.z; m = m > 0.f ? m : NEG_SLOPE * m; sum += at.z * m;
  m = xls.w + xrd.w + la * we.w; m = m > 0.f ? m : NEG_SLOPE * m; sum += at.w * m;
  sum += __shfl_xor(sum, 1, 32);
  sum += __shfl_xor(sum, 2, 32);
  sum += __shfl_xor(sum, 4, 32);
  if ((lane & 7) == 0) {
    int h = lane >> 3;
    sloop[(size_t)n * 4 + h] = sum;
    smax_u[(size_t)n * 4 + h] = enc_f(sum);
  }
}

// ---------------------------------------------------------------------------
// Edge score: one wave per edge; atomicMax into per-(dst,head) smax.
// ---------------------------------------------------------------------------
__global__ __launch_bounds__(256) void edge_score_kernel(
    const float* __restrict__ xl, const float* __restrict__ xr,
    const long long* __restrict__ ei, const float* __restrict__ ea,
    const float* __restrict__ We, const float* __restrict__ att,
    float* __restrict__ score, unsigned* __restrict__ smax_u, int E) {
  int e = (int)((blockIdx.x * (size_t)blockDim.x + threadIdx.x) >> 5);
  int lane = threadIdx.x & 31;
  if (e >= E) return;
  int s = (int)ei[e], d = (int)ei[(size_t)E + e];
  if (s == d) {  // pre-existing self loop: removed -> -inf score
    if (lane < 4) score[(size_t)e * 4 + lane] = -__builtin_inff();
    return;
  }
  int base = lane * 4;
  float4 xls = *(const float4*)(xl + (size_t)s * 128 + base);
  float4 xrd = *(const float4*)(xr + (size_t)d * 128 + base);
  float4 we = *(const float4*)(We + base);
  float4 at = *(const float4*)(att + base);
  float a = ea[e];
  float m, sum = 0.f;
  m = xls.x + xrd.x + a * we.x; m = m > 0.f ? m : NEG_SLOPE * m; sum += at.x * m;
  m = xls.y + xrd.y + a * we.y; m = m > 0.f ? m : NEG_SLOPE * m; sum += at.y * m;
  m = xls.z + xrd.z + a * we.z; m = m > 0.f ? m : NEG_SLOPE * m; sum += at.z * m;
  m = xls.w + xrd.w + a * we.w; m = m > 0.f ? m : NEG_SLOPE * m; sum += at.w * m;
  sum += __shfl_xor(sum, 1, 32);
  sum += __shfl_xor(sum, 2, 32);
  sum += __shfl_xor(sum, 4, 32);
  if ((lane & 7) == 0) {
    int h = lane >> 3;
    score[(size_t)e * 4 + h] = sum;
    atomicMax(&smax_u[(size_t)d * 4 + h], enc_f(sum));
  }
}

// Decode smax in place; denom = exp(score_loop - smax) (self-loop term).
__global__ void decode_denom_kernel(unsigned* __restrict__ smax_u,
                                    const float* __restrict__ sloop,
                                    float* __restrict__ denom, int n4) {
  int i = blockIdx.x * blockDim.x + threadIdx.x;
  if (i >= n4) return;
  float m = dec_f(smax_u[i]);
  ((float*)smax_u)[i] = m;
  denom[i] = expf(sloop[i] - m);
}

// Per (edge, head): denom[dst,h] += exp(score - smax[dst,h]).
__global__ void edge_denom_kernel(const float* __restrict__ score,
                                  const long long* __restrict__ ei,
                                  const float* __restrict__ smax,
                                  float* __restrict__ denom, int E) {
  int i = blockIdx.x * blockDim.x + threadIdx.x;
  if (i >= E * 4) return;
  float sc = score[i];
  if (sc == -__builtin_inff()) return;
  int e = i >> 2, h = i & 3;
  int d = (int)ei[(size_t)E + e];
  atomicAdd(&denom[(size_t)d * 4 + h], expf(sc - smax[(size_t)d * 4 + h]));
}

// acc = alpha_loop * xl (self-loop contribution).
__global__ void out_init_kernel(const float* __restrict__ xl,
                                const float* __restrict__ sloop,
                                const float* __restrict__ smax,
                                const float* __restrict__ denom,
                                float* __restrict__ acc, int N) {
  size_t i = blockIdx.x * (size_t)blockDim.x + threadIdx.x;
  if (i >= (size_t)N * 128) return;
  int n = (int)(i >> 7);
  int n4 = n * 4 + (int)((i & 127) >> 5);
  float al = expf(sloop[n4] - smax[n4]) / denom[n4];
  acc[i] = al * xl[i];
}

// One wave per edge: acc[dst] += alpha * xl[src]  (4 floats per lane).
__global__ __launch_bounds__(256) void aggregate_kernel(
    const float* __restrict__ xl, const long long* __restrict__ ei,
    const float* __restrict__ score, const float* __restrict__ smax,
    const float* __restrict__ denom, float* __restrict__ acc, int E) {
  int e = (int)((blockIdx.x * (size_t)blockDim.x + threadIdx.x) >> 5);
  int lane = threadIdx.x & 31;
  if (e >= E) return;
  int s = (int)ei[e], d = (int)ei[(size_t)E + e];
  if (s == d) return;
  int h = lane >> 3;
  size_t n4 = (size_t)d * 4 + h;
  float al = expf(score[(size_t)e * 4 + h] - smax[n4]) / denom[n4];
  int base = lane * 4;
  float4 xls = *(const float4*)(xl + (size_t)s * 128 + base);
  float* p = acc + (size_t)d * 128 + base;
  atomicAdd(p + 0, al * xls.x);
  atomicAdd(p + 1, al * xls.y);
  atomicAdd(p + 2, al * xls.z);
  atomicAdd(p + 3, al * xls.w);
}

// h = elu(acc + bias), in place.
__global__ void finalize_kernel(float* __restrict__ acc,
                                const float* __restrict__ b, int N) {
  size_t i = blockIdx.x * (size_t)blockDim.x + threadIdx.x;
  if (i >= (size_t)N * 128) return;
  float v = acc[i] + b[i & 127];
  acc[i] = v > 0.f ? v : expm1f(v);
}

// ---------------------------------------------------------------------------
static void run_layer(const float* Xin, const float* Wl, const float* bl,
                      const float* Wr, const float* br, const float* We,
                      const float* att, const float* bvec,
                      const long long* ei, const float* ea, float* xl, float* xr,
                      float* acc, float* score, float* sloop, float* smax,
                      float* denom, const float* sums, const float* cnts, int N,
                      int E, hipStream_t stream) {
  dim3 blk(256);
  int rowTiles = N / 16;  // N = 20000 is divisible by 16
  gemm_k128_wmma<8><<<(rowTiles + 7) / 8, blk, 0, stream>>>(Xin, Wl, bl, xl, N);
  gemm_k128_wmma<8><<<(rowTiles + 7) / 8, blk, 0, stream>>>(Xin, Wr, br, xr, N);
  node_score_kernel<<<(N + 7) / 8, blk, 0, stream>>>(
      xl, xr, We, att, sums, cnts, sloop, (unsigned*)smax, N);
  edge_score_kernel<<<(E + 7) / 8, blk, 0, stream>>>(
      xl, xr, ei, ea, We, att, score, (unsigned*)smax, E);
  decode_denom_kernel<<<(N * 4 + 255) / 256, blk, 0, stream>>>(
      (unsigned*)smax, sloop, denom, N * 4);
  edge_denom_kernel<<<(E * 4 + 255) / 256, blk, 0, stream>>>(score, ei, smax,
                                                             denom, E);
  out_init_kernel<<<(int)(((size_t)N * 128 + 255) / 256), blk, 0, stream>>>(
      xl, sloop, smax, denom, acc, N);
  aggregate_kernel<<<(E + 7) / 8, blk, 0, stream>>>(xl, ei, score, smax, denom,
                                                    acc, E);
  finalize_kernel<<<(int)(((size_t)N * 128 + 255) / 256), blk, 0, stream>>>(
      acc, bvec, N);
}

extern "C" void kernel_launch(void* const* d_in, const int* in_sizes, int n_in,
                              void* d_out, int out_size, void* d_ws,
                              size_t ws_size, hipStream_t stream) {
  (void)n_in; (void)out_size; (void)ws_size;
  const float* x = (const float*)d_in[0];
  const long long* ei = (const long long*)d_in[1];
  const float* ea = (const float*)d_in[2];
  const float* Wl1 = (const float*)d_in[3];
  const float* bl1 = (const float*)d_in[4];
  const float* Wr1 = (const float*)d_in[5];
  const float* br1 = (const float*)d_in[6];
  const float* We1 = (const float*)d_in[7];
  const float* att1 = (const float*)d_in[8];
  const float* b1 = (const float*)d_in[9];
  const float* Wl2 = (const float*)d_in[10];
  const float* bl2 = (const float*)d_in[11];
  const float* Wr2 = (const float*)d_in[12];
  const float* br2 = (const float*)d_in[13];
  const float* We2 = (const float*)d_in[14];
  const float* att2 = (const float*)d_in[15];
  const float* b2 = (const float*)d_in[16];
  const float* Wo = (const float*)d_in[17];
  const float* bo = (const float*)d_in[18];

  int N = in_sizes[0] / 128;
  int E = in_sizes[2];

  // Workspace layout (floats): ~42 MB total.
  float* ws = (float*)d_ws;
  size_t NH = (size_t)N * 128;
  float* xl = ws;
  float* xr = xl + NH;
  float* acc = xr + NH;
  float* score = acc + NH;                // E*4
  float* sloop = score + (size_t)E * 4;   // N*4
  float* smax = sloop + (size_t)N * 4;    // N*4 (uint-punned during max phase)
  float* denom = smax + (size_t)N * 4;    // N*4
  float* sums = denom + (size_t)N * 4;    // N
  float* cnts = sums + N;                 // N

  dim3 blk(256);
  zero2_kernel<<<(N + 255) / 256, blk, 0, stream>>>(sums, cnts, N);
  loopattr_edge_kernel<<<(E + 255) / 256, blk, 0, stream>>>(ei, ea, sums, cnts,
                                                            E);

  run_layer(x, Wl1, bl1, Wr1, br1, We1, att1, b1, ei, ea, xl, xr, acc, score,
            sloop, smax, denom, sums, cnts, N, E, stream);
  run_layer(acc, Wl2, bl2, Wr2, br2, We2, att2, b2, ei, ea, xl, xr, acc, score,
            sloop, smax, denom, sums, cnts, N, E, stream);

  // Final projection: [N x 128] @ [128 x 64] + bo -> d_out.
  gemm_k128_wmma<4><<<((N / 16) + 7) / 8, blk, 0, stream>>>(acc, Wo, bo,
                                                            (float*)d_out, N);
}